// IMDB_FCN_7430293422287
// MI455X (gfx1250) — compile-verified
//
#include <hip/hip_runtime.h>

typedef float v2f __attribute__((ext_vector_type(2)));
typedef float v8f __attribute__((ext_vector_type(8)));

#define BATCH 256
#define SEQ   2048
#define VOCAB 100000
#define EMB   128

// ---------------------------------------------------------------------------
// Kernel A: w[v] = dot(emb[v, :], fc_w[:])   (LUT build, [V,128] x [128,1])
// One wave handles 16 vocab rows via V_WMMA_F32_16X16X4_F32, looping K in
// steps of 4.  fc_w chunk is replicated into all 16 B-columns so every D
// column carries the same 16 dot products.
// A layout (ISA 7.12.2, 32-bit A 16x4): lane L holds row M=L%16, and the
// K-pair starting at K = 2*(L/16); VGPR0=K even, VGPR1=K odd.  B mirrors it.
// ---------------------------------------------------------------------------
__global__ __launch_bounds__(256) void vocab_dot_wmma(const float* __restrict__ emb,
                                                      const float* __restrict__ fcw,
                                                      float* __restrict__ w) {
    const int lane  = threadIdx.x & 31;
    const int wave  = (int)(blockIdx.x * (blockDim.x >> 5) + (threadIdx.x >> 5));
    const int nwave = VOCAB / 16;           // 6250, exact
    if (wave >= nwave) return;              // wave-uniform: EXEC stays all-ones

    const int rowBase = wave * 16;
    const int m       = lane & 15;          // row within the 16-row tile
    const int kh      = (lane >> 4) << 1;   // 0 for lanes 0-15, 2 for lanes 16-31
    const float* rowPtr = emb + (size_t)(rowBase + m) * EMB;

#if __has_builtin(__builtin_amdgcn_wmma_f32_16x16x4_f32)
    v8f c = {0.f, 0.f, 0.f, 0.f, 0.f, 0.f, 0.f, 0.f};
#pragma unroll
    for (int k = 0; k < EMB; k += 4) {
        v2f a = *(const v2f*)(rowPtr + k + kh);   // A[m, k+kh : k+kh+2)
        v2f b = *(const v2f*)(fcw    + k + kh);   // B[k+kh : k+kh+2, all N]
        // 8 args: (neg_a, A, neg_b, B, c_mod, C, reuse_a, reuse_b)
        c = __builtin_amdgcn_wmma_f32_16x16x4_f32(
                false, a, false, b, (short)0, c, false, false);
    }
    // All D columns identical.  C/D layout: lanes 0-15 hold M=0..7 in c[0..7],
    // lanes 16-31 hold M=8..15.  Lanes 0 and 16 write the 16 results.
    if (m == 0) {
        float* dst = w + rowBase + (lane >> 4) * 8;
#pragma unroll
        for (int i = 0; i < 8; ++i) dst[i] = c[i];
    }
#else
    // VALU fallback: 1 row per lane (lanes 0-15), plain FMA chain.
    for (int r = lane; r < 16; r += 32) {
        const float* rp = emb + (size_t)(rowBase + r) * EMB;
        float acc = 0.f;
#pragma unroll
        for (int k = 0; k < EMB; ++k) acc += rp[k] * fcw[k];
        w[rowBase + r] = acc;
    }
#endif
}

// ---------------------------------------------------------------------------
// Kernel B: one 256-thread block per batch row.
//   out[b] = (sum_{s<len} w[text[b,s]]) / len + fc_b
// text loads are fully coalesced; w gathers hit L2 (400KB LUT).
// ---------------------------------------------------------------------------
__global__ __launch_bounds__(256) void pool_reduce(const int* __restrict__ text,
                                                   const int* __restrict__ lens,
                                                   const float* __restrict__ w,
                                                   const float* __restrict__ fcb,
                                                   float* __restrict__ out) {
    __shared__ float sdata[256];
    const int b   = blockIdx.x;
    const int tid = threadIdx.x;
    const int len = lens[b];
    const int* row = text + (size_t)b * SEQ;

    float sum = 0.f;
#pragma unroll
    for (int s = tid; s < SEQ; s += 256) {
        if (s < len) sum += w[row[s]];
    }
    sdata[tid] = sum;
    __syncthreads();
#pragma unroll
    for (int off = 128; off > 0; off >>= 1) {
        if (tid < off) sdata[tid] += sdata[tid + off];
        __syncthreads();
    }
    if (tid == 0) out[b] = sdata[0] / (float)len + fcb[0];
}

// ---------------------------------------------------------------------------
extern "C" void kernel_launch(void* const* d_in, const int* in_sizes, int n_in,
                              void* d_out, int out_size, void* d_ws, size_t ws_size,
                              hipStream_t stream) {
    (void)in_sizes; (void)n_in; (void)out_size; (void)ws_size;
    const int*   text = (const int*)  d_in[0];   // [B, S]
    const int*   lens = (const int*)  d_in[1];   // [B]
    const float* emb  = (const float*)d_in[2];   // [V, E]
    const float* fcw  = (const float*)d_in[3];   // [1, E]
    const float* fcb  = (const float*)d_in[4];   // [1]
    float*       out  = (float*)d_out;           // [B, 1]
    float*       w    = (float*)d_ws;            // V floats (400 KB) scratch LUT

    const int nwave   = VOCAB / 16;              // 6250 waves, 16 rows each
    const int blocksA = (nwave + 7) / 8;         // 8 waves (256 thr) per block

    vocab_dot_wmma<<<blocksA, 256, 0, stream>>>(emb, fcw, w);
    pool_reduce   <<<BATCH,   256, 0, stream>>>(text, lens, w, fcb, out);
}